// Simulate_22497038696790
// MI455X (gfx1250) — compile-verified
//
#include <hip/hip_runtime.h>
#include <hip/hip_bf16.h>

#define NSUB 64

// RHS of the RSV target-cell-limited model (thr state is identically 0, dropped).
__device__ __forceinline__ void rsv_rhs(float be, float de, float pp, float cc,
                                        float D, float E, float V,
                                        float& dD, float& dE, float& dV) {
    float T = 1.0f - D - E;          // remaining target-cell fraction
    dD = de * E;                     // delta * E
    dE = fmaf(be * V, T, -dD);       // beta*V*T - delta*E
    dV = fmaf(pp, E, -(cc * V));     // p*E - c*V
}

__device__ __forceinline__ void rk4_step(float be, float de, float pp, float cc,
                                         float dt, float& D, float& E, float& V) {
    float k1D, k1E, k1V, k2D, k2E, k2V, k3D, k3E, k3V, k4D, k4E, k4V;
    rsv_rhs(be, de, pp, cc, D, E, V, k1D, k1E, k1V);
    float h = dt * 0.5f;
    rsv_rhs(be, de, pp, cc, fmaf(h, k1D, D), fmaf(h, k1E, E), fmaf(h, k1V, V),
            k2D, k2E, k2V);
    rsv_rhs(be, de, pp, cc, fmaf(h, k2D, D), fmaf(h, k2E, E), fmaf(h, k2V, V),
            k3D, k3E, k3V);
    rsv_rhs(be, de, pp, cc, fmaf(dt, k3D, D), fmaf(dt, k3E, E), fmaf(dt, k3V, V),
            k4D, k4E, k4V);
    float sixth = dt * (1.0f / 6.0f);
    D = fmaf(sixth, k1D + 2.0f * (k2D + k3D) + k4D, D);
    E = fmaf(sixth, k1E + 2.0f * (k2E + k3E) + k4E, E);
    V = fmaf(sixth, k1V + 2.0f * (k2V + k3V) + k4V, V);
}

// One wave per workgroup (wave32) -> 512 workgroups for B=16384, spreading the
// limited wave count across as many WGPs/SIMD32s as possible.
__global__ __launch_bounds__(32)
void rsv_rk4_kernel(const float* __restrict__ days1, const float* __restrict__ days2,
                    const float* __restrict__ D0,    const float* __restrict__ E0,
                    const float* __restrict__ V01,   const float* __restrict__ V02,
                    const float* __restrict__ beta,  const float* __restrict__ delta,
                    const float* __restrict__ p,     const float* __restrict__ c,
                    const int*   __restrict__ treatment,
                    float* __restrict__ out, int B, int nt)
{
    int b = blockIdx.x * blockDim.x + threadIdx.x;
    if (b >= B) return;

    const float be = beta[b], de = delta[b], pp = p[b], cc = c[b];
    const int   tr = treatment[0];

    float D = D0[b], E = E0[b];
    float V = tr ? V01[b] : 0.0f;

    // argmax(days1 == 15.0): first matching index, 0 if none.
    int idx15 = 0;
    for (int t = nt - 1; t >= 0; --t)
        if (days1[t] == 15.0f) idx15 = t;

    float sD = D, sE = E, sV = V;

    const size_t planeB = (size_t)B;
    const size_t comp   = (size_t)nt * planeB;   // stride between components
    float* o1 = out;                             // phase 0 base: [4, nt, B]
    float* o2 = out + 4 * comp;                  // phase 1 base

    // ---- Phase 1: integrate from t0 = 5.0 over days1 ----
    float tprev = 5.0f;
    for (int t = 0; t < nt; ++t) {
        float tn = days1[t];
        float dt = (tn - tprev) * (1.0f / NSUB); // /64 == *2^-6, exact
        tprev = tn;
        if (dt != 0.0f) {                        // dt==0 segment is an exact no-op
            #pragma unroll 4
            for (int s = 0; s < NSUB; ++s)
                rk4_step(be, de, pp, cc, dt, D, E, V);
        }
        size_t o = (size_t)t * planeB + (size_t)b;
        o1[o]            = D;
        o1[comp + o]     = E;
        o1[2 * comp + o] = V;
        o1[3 * comp + o] = __logf(V);            // v_log_f32 path
        if (t == idx15) { sD = D; sE = E; sV = V; }
    }

    // ---- Phase 2: restart from saved day-15 state, add V02 under treatment ----
    D = sD; E = sE; V = tr ? (sV + V02[b]) : sV;
    tprev = 15.0f;
    for (int t = 0; t < nt; ++t) {
        float tn = days2[t];
        float dt = (tn - tprev) * (1.0f / NSUB);
        tprev = tn;
        if (dt != 0.0f) {
            #pragma unroll 4
            for (int s = 0; s < NSUB; ++s)
                rk4_step(be, de, pp, cc, dt, D, E, V);
        }
        size_t o = (size_t)t * planeB + (size_t)b;
        o2[o]            = D;
        o2[comp + o]     = E;
        o2[2 * comp + o] = V;
        o2[3 * comp + o] = __logf(V);
    }
}

extern "C" void kernel_launch(void* const* d_in, const int* in_sizes, int n_in,
                              void* d_out, int out_size, void* d_ws, size_t ws_size,
                              hipStream_t stream) {
    // setup_inputs() order:
    // 0: days1[17]  1: days2[17]  2: D0[B]  3: E0[B]  4: V01[B]  5: V02[B]
    // 6: beta[B]    7: delta[B]   8: p[B]   9: c[B]   10: treatment (int, 1 elem)
    const float* days1     = (const float*)d_in[0];
    const float* days2     = (const float*)d_in[1];
    const float* D0        = (const float*)d_in[2];
    const float* E0        = (const float*)d_in[3];
    const float* V01       = (const float*)d_in[4];
    const float* V02       = (const float*)d_in[5];
    const float* beta      = (const float*)d_in[6];
    const float* delta     = (const float*)d_in[7];
    const float* p         = (const float*)d_in[8];
    const float* c         = (const float*)d_in[9];
    const int*   treatment = (const int*)d_in[10];
    float*       out       = (float*)d_out;

    const int B  = in_sizes[2];   // batch size (16384)
    const int nt = in_sizes[0];   // save points per phase (17)

    const int block = 32;                       // one wave32 per workgroup
    const int grid  = (B + block - 1) / block;  // 512 workgroups
    rsv_rk4_kernel<<<grid, block, 0, stream>>>(days1, days2, D0, E0, V01, V02,
                                               beta, delta, p, c, treatment,
                                               out, B, nt);
}